// MolGDL_11158325035411
// MI455X (gfx1250) — compile-verified
//
#include <hip/hip_runtime.h>
#include <hip/hip_bf16.h>
#include <cstdint>

typedef __attribute__((ext_vector_type(2))) float v2f;
typedef __attribute__((ext_vector_type(8))) float v8f;

#define N_NODES   20000
#define N_EDGES   320000
#define IN_FEAT   128
#define HIDDEN    256
#define OUT_DIM   12
#define NUM_SCALES 5
#define NUM_CONVS  3
#define NUM_GRAPHS 800
#define BN_EPS    1e-5f

// GEMM register blocking: each wave computes MBLK x NBLK tiles of 16x16
#define MBLK 5
#define NBLK 2

// ---------------------------------------------------------------- utilities
__global__ void zero_f32(float* __restrict__ p, int n) {
    int i = blockIdx.x * blockDim.x + threadIdx.x;
    if (i < n) p[i] = 0.0f;
}

__global__ void deg_accum(const int* __restrict__ dst, float* __restrict__ degs) {
    int i = blockIdx.x * blockDim.x + threadIdx.x;
    if (i < N_EDGES) atomicAdd(&degs[dst[i]], 1.0f);
}

__global__ void count_accum(const int* __restrict__ gid, float* __restrict__ counts) {
    int i = blockIdx.x * blockDim.x + threadIdx.x;
    if (i < N_NODES) atomicAdd(&counts[gid[i]], 1.0f);
}

__global__ void make_norm(const float* __restrict__ degs, float* __restrict__ norm) {
    int i = blockIdx.x * blockDim.x + threadIdx.x;
    if (i < N_NODES) norm[i] = rsqrtf(fmaxf(degs[i], 1.0f));
}

// --------------------------------------------------- edge scatter (hn[src] -> agg[dst])
__global__ void edge_scatter(const float* __restrict__ h, const float* __restrict__ norm,
                             const int* __restrict__ src, const int* __restrict__ dst,
                             float* __restrict__ agg) {
    int idx = blockIdx.x * blockDim.x + threadIdx.x;
    int e = idx >> 6;                       // HIDDEN/4 = 64 chunks per edge
    if (e >= N_EDGES) return;
    int c = (idx & 63) << 2;
    int s = src[e], d = dst[e];
    float ns = norm[s];
    float4 v = *(const float4*)(h + (size_t)s * HIDDEN + c);
    float* base = agg + (size_t)d * HIDDEN + c;
    atomicAdd(base + 0, v.x * ns);
    atomicAdd(base + 1, v.y * ns);
    atomicAdd(base + 2, v.z * ns);
    atomicAdd(base + 3, v.w * ns);
}

// --------------------------------------------------- graph pooling accumulate
__global__ void pool_accum(const float* __restrict__ h, const int* __restrict__ gid,
                           float* __restrict__ pool) {
    int idx = blockIdx.x * blockDim.x + threadIdx.x;
    int n = idx >> 6;
    if (n >= N_NODES) return;
    int c = (idx & 63) << 2;
    int g = gid[n];
    float4 v = *(const float4*)(h + (size_t)n * HIDDEN + c);
    float* base = pool + (size_t)g * HIDDEN + c;
    atomicAdd(base + 0, v.x);
    atomicAdd(base + 1, v.y);
    atomicAdd(base + 2, v.z);
    atomicAdd(base + 3, v.w);
}

// --------------------------------------------------- fp32 WMMA GEMM (K, N compile-time)
// C = act(bn(rowScale.A @ B) + colBias),  N = HIDDEN = 256 fixed.
// block = 128 threads = 4 waves; wave computes 80x32 macro-tile (5x2 tiles of 16x16).
// grid.x = M/80, grid.y = N/128.  M multiple of 80, K multiple of 4.
// All tile strides are compile-time -> loads use immediate offsets off 2 base pointers.
template <int K>
__global__ void wmma_gemm_f32(const float* __restrict__ A, const float* __restrict__ B,
                              float* __restrict__ C,
                              const float* __restrict__ rowScale,
                              const float* __restrict__ bnG, const float* __restrict__ bnB,
                              const float* __restrict__ bnM, const float* __restrict__ bnV,
                              const float* __restrict__ colBias, int act) {
    constexpr int N = HIDDEN;
    const int lane = threadIdx.x & 31;
    const int wave = threadIdx.x >> 5;
    const int m0 = blockIdx.x * (16 * MBLK);
    const int n0 = (blockIdx.y * 4 + wave) * (16 * NBLK);

    const int lr = lane & 15;
    const int hi = lane >> 4;            // 0: lanes 0-15, 1: lanes 16-31
    const int kb = hi * 2;               // K sub-offset within the 4-step

    // base pointers; all per-tile / per-k strides are compile-time immediates
    const float* ap = A + (size_t)(m0 + lr) * K + kb;   // A tile t: ap + t*16*K + k
    const float* bp = B + (size_t)kb * N + (n0 + lr);   // B row k+kb, col j: bp + k*N + j*16

    float rs[MBLK];
#pragma unroll
    for (int t = 0; t < MBLK; ++t)
        rs[t] = rowScale ? rowScale[m0 + t * 16 + lr] : 1.0f;

    v8f acc[MBLK * NBLK];
#pragma unroll
    for (int i = 0; i < MBLK * NBLK; ++i) acc[i] = (v8f){0.f,0.f,0.f,0.f,0.f,0.f,0.f,0.f};

#pragma unroll 2
    for (int k = 0; k < K; k += 4) {
        v2f bfr[NBLK];
#pragma unroll
        for (int j = 0; j < NBLK; ++j) {
            bfr[j].x = bp[(size_t)k * N + j * 16];           // row k+kb
            bfr[j].y = bp[(size_t)(k + 1) * N + j * 16];     // row k+kb+1
        }
#pragma unroll
        for (int t = 0; t < MBLK; ++t) {
            v2f a = *(const v2f*)(ap + t * 16 * K + k);
            a.x *= rs[t]; a.y *= rs[t];
#pragma unroll
            for (int j = 0; j < NBLK; ++j) {
                acc[t * NBLK + j] = __builtin_amdgcn_wmma_f32_16x16x4_f32(
                    false, a, false, bfr[j], (short)0, acc[t * NBLK + j], false, false);
            }
        }
    }

    // fused column-wise affine (BN folded / bias) + activation
    float scl[NBLK], bias[NBLK];
#pragma unroll
    for (int j = 0; j < NBLK; ++j) {
        int n = n0 + j * 16 + lr;
        scl[j] = 1.0f; bias[j] = 0.0f;
        if (bnG) {
            float inv = rsqrtf(bnV[n] + BN_EPS);
            scl[j]  = bnG[n] * inv;
            bias[j] = bnB[n] - bnM[n] * scl[j];
        } else if (colBias) {
            bias[j] = colBias[n];
        }
    }
    float* cp = C + (size_t)(m0 + hi * 8) * N + n0 + lr;    // row m0+hi*8, col n0+lr
#pragma unroll
    for (int t = 0; t < MBLK; ++t) {
#pragma unroll
        for (int j = 0; j < NBLK; ++j) {
#pragma unroll
            for (int v = 0; v < 8; ++v) {
                float x = acc[t * NBLK + j][v] * scl[j] + bias[j];
                if (act == 1)      x = fmaxf(x, 0.0f);              // ReLU
                else if (act == 2) x = (x > 0.0f) ? x : 0.01f * x;  // LeakyReLU
                cp[(size_t)(t * 16 + v) * N + j * 16] = x;
            }
        }
    }
}

// --------------------------------------------------- pool -> mean over scales -> L2 normalize
__global__ void finalize_pool(const float* __restrict__ pool, const float* __restrict__ counts,
                              float* __restrict__ hnorm) {
    __shared__ float red[HIDDEN];
    int g = blockIdx.x, t = threadIdx.x;
    float c = fmaxf(counts[g], 1.0f);
    float v = pool[(size_t)g * HIDDEN + t] / (c * (float)NUM_SCALES);
    red[t] = v * v;
    __syncthreads();
    for (int s = HIDDEN / 2; s > 0; s >>= 1) {
        if (t < s) red[t] += red[t + s];
        __syncthreads();
    }
    float nrm = fmaxf(sqrtf(red[0]), 1e-12f);
    hnorm[(size_t)g * HIDDEN + t] = v / nrm;
}

// --------------------------------------------------- output head: sigmoid(h2 @ W_out + b_out)
__global__ void out_head(const float* __restrict__ h2, const float* __restrict__ W,
                         const float* __restrict__ b, float* __restrict__ out) {
    int idx = blockIdx.x * blockDim.x + threadIdx.x;
    if (idx >= NUM_GRAPHS * OUT_DIM) return;
    int g = idx / OUT_DIM, o = idx % OUT_DIM;
    float acc = b[o];
    const float* hr = h2 + (size_t)g * HIDDEN;
#pragma unroll 8
    for (int k = 0; k < HIDDEN; ++k) acc += hr[k] * W[k * OUT_DIM + o];
    out[idx] = 1.0f / (1.0f + __expf(-acc));
}

// ================================================================ launch
extern "C" void kernel_launch(void* const* d_in, const int* in_sizes, int n_in,
                              void* d_out, int out_size, void* d_ws, size_t ws_size,
                              hipStream_t stream) {
    const float* features = (const float*)d_in[0];
    const int*   src      = (const int*)  d_in[1];
    const int*   dst      = (const int*)  d_in[2];
    const int*   gid      = (const int*)  d_in[3];
    // d_in[4] = num_graphs scalar (compile-time constant here)
    const float* W_proj   = (const float*)d_in[5];
    const float* W_conv   = (const float*)d_in[6];
    const float* bnG      = (const float*)d_in[7];
    const float* bnB      = (const float*)d_in[8];
    const float* bnM      = (const float*)d_in[9];
    const float* bnV      = (const float*)d_in[10];
    const float* W_fc1    = (const float*)d_in[11];
    const float* b_fc1    = (const float*)d_in[12];
    const float* W_out    = (const float*)d_in[13];
    const float* b_out    = (const float*)d_in[14];
    float* out = (float*)d_out;

    // workspace layout (all 16-float aligned)
    float* ws     = (float*)d_ws;
    float* degs   = ws;                                       // 20000 (pad 20480)
    float* norm   = degs   + 20480;                           // 20000 (pad 20480)
    float* counts = norm   + 20480;                           // 800  (pad 1024)
    float* h      = counts + 1024;                            // N_NODES*HIDDEN
    float* agg    = h      + (size_t)N_NODES * HIDDEN;        // N_NODES*HIDDEN
    float* pool   = agg    + (size_t)N_NODES * HIDDEN;        // NUM_GRAPHS*HIDDEN
    float* hnorm  = pool   + (size_t)NUM_GRAPHS * HIDDEN;     // NUM_GRAPHS*HIDDEN
    float* h2     = hnorm  + (size_t)NUM_GRAPHS * HIDDEN;     // NUM_GRAPHS*HIDDEN

    const int TPB = 256;
    auto cdiv = [](int a, int b) { return (a + b - 1) / b; };

    // degree / counts / pooling accumulators
    zero_f32<<<cdiv(N_NODES, TPB), TPB, 0, stream>>>(degs, N_NODES);
    zero_f32<<<cdiv(NUM_GRAPHS, TPB), TPB, 0, stream>>>(counts, NUM_GRAPHS);
    zero_f32<<<cdiv(NUM_GRAPHS * HIDDEN, TPB), TPB, 0, stream>>>(pool, NUM_GRAPHS * HIDDEN);
    deg_accum<<<cdiv(N_EDGES, TPB), TPB, 0, stream>>>(dst, degs);
    count_accum<<<cdiv(N_NODES, TPB), TPB, 0, stream>>>(gid, counts);
    make_norm<<<cdiv(N_NODES, TPB), TPB, 0, stream>>>(degs, norm);

    const dim3 gemmBlock(128);                                              // 4 waves
    const dim3 gridProj(N_NODES / (16 * MBLK), HIDDEN / (4 * 16 * NBLK));   // 250 x 2
    const dim3 gridFc(NUM_GRAPHS / (16 * MBLK), HIDDEN / (4 * 16 * NBLK)); // 10 x 2

    for (int s = 0; s < NUM_SCALES; ++s) {
        // h = features @ W_proj[s]
        wmma_gemm_f32<IN_FEAT><<<gridProj, gemmBlock, 0, stream>>>(
            features, W_proj + (size_t)s * IN_FEAT * HIDDEN, h,
            nullptr, nullptr, nullptr, nullptr, nullptr, nullptr, 0);

        for (int i = 0; i < NUM_CONVS; ++i) {
            const int li = s * NUM_CONVS + i;
            zero_f32<<<cdiv(N_NODES * HIDDEN, TPB), TPB, 0, stream>>>(agg, N_NODES * HIDDEN);
            edge_scatter<<<cdiv(N_EDGES * (HIDDEN / 4), TPB), TPB, 0, stream>>>(
                h, norm, src, dst, agg);
            // h = relu(bn((agg * norm[dst]) @ W_conv[s,i]))
            wmma_gemm_f32<HIDDEN><<<gridProj, gemmBlock, 0, stream>>>(
                agg, W_conv + (size_t)li * HIDDEN * HIDDEN, h,
                norm,
                bnG + (size_t)li * HIDDEN, bnB + (size_t)li * HIDDEN,
                bnM + (size_t)li * HIDDEN, bnV + (size_t)li * HIDDEN,
                nullptr, 1);
        }
        pool_accum<<<cdiv(N_NODES * (HIDDEN / 4), TPB), TPB, 0, stream>>>(h, gid, pool);
    }

    // mean over scales, per-graph mean, L2 row normalize
    finalize_pool<<<NUM_GRAPHS, HIDDEN, 0, stream>>>(pool, counts, hnorm);

    // h2 = leaky_relu(hnorm @ W_fc1 + b_fc1)
    wmma_gemm_f32<HIDDEN><<<gridFc, gemmBlock, 0, stream>>>(
        hnorm, W_fc1, h2,
        nullptr, nullptr, nullptr, nullptr, nullptr, b_fc1, 2);

    // out = sigmoid(h2 @ W_out + b_out)
    out_head<<<cdiv(NUM_GRAPHS * OUT_DIM, TPB), TPB, 0, stream>>>(h2, W_out, b_out, out);
}